// NTN_45432164057146
// MI455X (gfx1250) — compile-verified
//
#include <hip/hip_runtime.h>
#include <math.h>

typedef __attribute__((ext_vector_type(2))) float v2f;
typedef __attribute__((ext_vector_type(8))) float v8f;

#define NDIM     100
#define KSLICES  4
#define BATCH    4096
#define NREL     1000
#define MAX_ITEMS 512   // binomial(4096, 1/1000): P(n_r > 512) is effectively 0

// One block per relation r. 4 waves per block; wave w handles tensor slice k=w.
// Wave 0 builds the (deterministic, ascending) list of batch items with
// relations[b]==r via ballot-compaction into LDS. Each wave then processes the
// item list in chunks of 16 columns:
//   C(16x16) += Wk_rowblock(16x4) x E2cols(4x16)   via V_WMMA_F32_16X16X4_F32
// then contracts rows of C with e1 (per-column), giving g_b for 16 items.
__global__ __launch_bounds__(32 * KSLICES) void ntn_slice_kernel(
    const int* __restrict__ heads, const int* __restrict__ tails,
    const int* __restrict__ relations,
    const float* __restrict__ E,   // [N_ENTITIES, 100]
    const float* __restrict__ W,   // [1000, 4, 100, 100]
    const float* __restrict__ V,   // [1000, 4, 200]
    const float* __restrict__ Bp,  // [1000, 4]
    float* __restrict__ gsum)      // ws: [4096, 4] = g_a + g_b + bias
{
  __shared__ int items[MAX_ITEMS];
  __shared__ int s_count;

  const int r    = blockIdx.x;
  const int tid  = threadIdx.x;
  const int lane = tid & 31;
  const int wave = tid >> 5;

  if (wave == 0) {
    int cnt = 0;
    for (int base = 0; base < BATCH; base += 32) {
      const int b = base + lane;
      const bool match = (relations[b] == r);
      const unsigned long long m = __ballot(match);
      if (match) {
        const int pos = cnt + (int)__popcll(m & ((1ull << lane) - 1ull));
        if (pos < MAX_ITEMS) items[pos] = b;
      }
      cnt += (int)__popcll(m);
    }
    if (lane == 0) s_count = (cnt > MAX_ITEMS) ? MAX_ITEMS : cnt;
  }
  __syncthreads();

  const int count = s_count;
  if (count == 0) return;

  const int k = wave;
  const float* __restrict__ Wk = W + ((size_t)r * KSLICES + k) * (NDIM * NDIM);
  const float* __restrict__ Vk = V + ((size_t)r * KSLICES + k) * (2 * NDIM);
  const float  bias = Bp[r * KSLICES + k];

  const int col  = lane & 15;  // C/B column (batch item slot) AND A row-in-block
  const int half = lane >> 4;  // selects K-pair (A/B frags) and M-half (C frag)

  for (int cs = 0; cs < count; cs += 16) {
    const int sel  = cs + col;
    const int item = items[(sel < count) ? sel : (count - 1)];
    const float* __restrict__ e1p = E + (size_t)heads[item] * NDIM;
    const float* __restrict__ e2p = E + (size_t)tails[item] * NDIM;

    float gb = 0.0f;
    for (int I = 0; I < 7; ++I) {          // 7 row blocks of 16 cover 100 rows
      v8f c = {};
      const int  rowA = I * 16 + col;      // A-frag: M = lane&15 (+ block base)
      const bool rok  = (rowA < NDIM);
      const float* __restrict__ wrow = Wk + (size_t)rowA * NDIM;
      const v2f zero2 = {0.0f, 0.0f};
      #pragma unroll 5
      for (int s = 0; s < 25; ++s) {       // contraction: 25 steps of K=4
        const int cp = s * 4 + half * 2;   // K-pair for this lane half
        v2f a = rok ? *(const v2f*)(wrow + cp) : zero2;
        v2f b = *(const v2f*)(e2p + cp);
        c = __builtin_amdgcn_wmma_f32_16x16x4_f32(
            /*neg_a=*/false, a, /*neg_b=*/false, b,
            /*c_mod=*/(short)0, c, /*reuse_a=*/false, /*reuse_b=*/false);
      }
      // contract this row block's C with e1: lane holds rows rbase..rbase+7
      // of column `col` (C layout: VGPR j -> M = half*8 + j, N = lane&15)
      const int rbase = I * 16 + half * 8;
      #pragma unroll
      for (int j = 0; j < 8; ++j) {
        const int row = rbase + j;
        const float e1v = (row < NDIM) ? e1p[row] : 0.0f;
        gb += e1v * c[j];
      }
    }
    // combine the two M-halves of each column
    gb += __shfl_xor(gb, 16, 32);

    // g_a = v . [e1; e2]  (redundant in upper half lanes; only half 0 writes)
    float ga = 0.0f;
    for (int m = 0; m < NDIM; ++m) {
      ga = fmaf(Vk[m], e1p[m], ga);
      ga = fmaf(Vk[NDIM + m], e2p[m], ga);
    }

    if (half == 0 && sel < count) {
      gsum[(size_t)item * KSLICES + k] = ga + gb + bias;
    }
  }
}

__global__ void ntn_score_kernel(const int* __restrict__ relations,
                                 const float* __restrict__ U,
                                 const float* __restrict__ gsum,
                                 float* __restrict__ out) {
  const int b = blockIdx.x * blockDim.x + threadIdx.x;
  if (b >= BATCH) return;
  const int r = relations[b];
  float s = 0.0f;
  #pragma unroll
  for (int k = 0; k < KSLICES; ++k) {
    s = fmaf(U[r * KSLICES + k], tanhf(gsum[(size_t)b * KSLICES + k]), s);
  }
  out[b] = 1.0f / (1.0f + expf(-s));
}

extern "C" void kernel_launch(void* const* d_in, const int* in_sizes, int n_in,
                              void* d_out, int out_size, void* d_ws, size_t ws_size,
                              hipStream_t stream) {
  const int*   heads = (const int*)d_in[0];
  const int*   tails = (const int*)d_in[1];
  const int*   rels  = (const int*)d_in[2];
  const float* E     = (const float*)d_in[3];
  const float* W     = (const float*)d_in[4];
  const float* V     = (const float*)d_in[5];
  const float* Bp    = (const float*)d_in[6];
  const float* U     = (const float*)d_in[7];

  float* gsum = (float*)d_ws;  // [BATCH, KSLICES] — every slot written once

  ntn_slice_kernel<<<NREL, 32 * KSLICES, 0, stream>>>(
      heads, tails, rels, E, W, V, Bp, gsum);
  ntn_score_kernel<<<(BATCH + 255) / 256, 256, 0, stream>>>(
      rels, U, gsum, (float*)d_out);
}